// GSAPool_54425825575649
// MI455X (gfx1250) — compile-verified
//
#include <hip/hip_runtime.h>
#include <hip/hip_bf16.h>
#include <math.h>

// ---- problem constants (match reference) ----
#define NN   50000
#define EE   500000
#define DD   128
#define BB   50
#define NPG  1000
#define KSEL 500
#define NSEL (BB * KSEL)                 // 25000 selected nodes
#define XOUT_ELEMS ((size_t)NSEL * DD)   // 3,200,000 floats
#define ALPHA 0.6f

typedef float v2f __attribute__((ext_vector_type(2)));
typedef float v8f __attribute__((ext_vector_type(8)));
typedef int   v4i __attribute__((ext_vector_type(4)));

typedef __attribute__((address_space(1))) v4i* gv4i_ptr;
typedef __attribute__((address_space(3))) v4i* lv4i_ptr;

#if __has_builtin(__builtin_amdgcn_global_load_async_to_lds_b128) && \
    __has_builtin(__builtin_amdgcn_s_wait_asynccnt)
#define HAS_ASYNC_LDS 1
#else
#define HAS_ASYNC_LDS 0
#endif

// ---------------- utility fills ----------------
__global__ void k_fill_f(float* p, float v, int n) {
    int i = blockIdx.x * 256 + threadIdx.x;
    if (i < n) p[i] = v;
}
__global__ void k_fill_i(int* p, int v, int n) {
    int i = blockIdx.x * 256 + threadIdx.x;
    if (i < n) p[i] = v;
}

// ---------------- degree / dinv ----------------
__global__ void k_deg_edges(const int* __restrict__ dst, float* __restrict__ deg) {
    int e = blockIdx.x * 256 + threadIdx.x;
    if (e < EE) atomicAdd(&deg[dst[e]], 1.0f);
}
__global__ void k_dinv(float* deg) {
    int i = blockIdx.x * 256 + threadIdx.x;
    if (i < NN) deg[i] = rsqrtf(fmaxf(deg[i], 1e-12f));
}

// ---------------- scalar scores: xw_s = x@W_s ; score_f = x@W_f + b_f ----------------
__global__ __launch_bounds__(256) void k_scores(const float* __restrict__ x,
                                                const float* __restrict__ Ws,
                                                const float* __restrict__ Wf,
                                                const float* __restrict__ bf,
                                                float* __restrict__ xw_s,
                                                float* __restrict__ score_f) {
    int node = blockIdx.x * 8 + (threadIdx.x >> 5);
    int lane = threadIdx.x & 31;
    if (node >= NN) return;
    const float* row = x + (size_t)node * DD;
    float s1 = 0.f, s2 = 0.f;
    #pragma unroll
    for (int d = 0; d < DD; d += 32) {
        float v = row[d + lane];
        s1 += v * Ws[d + lane];
        s2 += v * Wf[d + lane];
    }
    for (int off = 16; off > 0; off >>= 1) {
        s1 += __shfl_down(s1, off, 32);
        s2 += __shfl_down(s2, off, 32);
    }
    if (lane == 0) { xw_s[node] = s1; score_f[node] = s2 + bf[0]; }
}

// ---------------- WMMA GEMM: xw_fu = x @ W_fu (f32, exact) ----------------
// Block: 256 threads = 8 waves; block computes 16 rows x 128 cols.
// Each wave: one 16x16 tile, K=128 as 32 x V_WMMA_F32_16X16X4_F32.
// W staged in 32-row chunks; async global->LDS (ASYNCcnt) double-buffered when available.

__device__ __forceinline__ void stage_w_async(const float* __restrict__ Wg,
                                              float* lds, int tid) {
#if HAS_ASYNC_LDS
    const float4* g = (const float4*)Wg;       // 32*128 floats = 1024 x b128
    float4*       l = (float4*)lds;
    #pragma unroll
    for (int j = 0; j < 4; ++j) {
        __builtin_amdgcn_global_load_async_to_lds_b128(
            (gv4i_ptr)(g + j * 256 + tid),
            (lv4i_ptr)(l + j * 256 + tid),
            0, 0);
    }
#endif
}

__global__ __launch_bounds__(256) void k_gemm_fu(const float* __restrict__ x,
                                                 const float* __restrict__ W,
                                                 float* __restrict__ xw) {
    __shared__ float sW[2][32][DD];  // 32 KB double-buffered K-chunks of W
    __shared__ float sA[16][132];    // padded: stride 132 -> no bank conflicts
    const int tid  = threadIdx.x;
    const int row0 = blockIdx.x * 16;

    // stage A slab (16 x 128) via float4
    {
        const float4* Xv = (const float4*)(x + (size_t)row0 * DD);
        for (int i = tid; i < 16 * DD / 4; i += 256) {
            int r = i >> 5, c4 = i & 31;
            *((float4*)&sA[r][c4 * 4]) = Xv[i];
        }
    }

    const int wave = tid >> 5, lane = tid & 31;
    const int half = lane >> 4, lm = lane & 15;
    const int col0 = wave * 16;
    v8f acc = {};

#if HAS_ASYNC_LDS
    // prologue: chunk 0 -> buffer 0
    stage_w_async(W, &sW[0][0][0], tid);
    #pragma unroll
    for (int c = 0; c < 4; ++c) {
        const int ko = 32 * c;
        if (c < 3) {
            stage_w_async(W + (size_t)(ko + 32) * DD, &sW[(c + 1) & 1][0][0], tid);
            __builtin_amdgcn_s_wait_asynccnt(4);   // oldest chunk (4 xfers/thread) done
        } else {
            __builtin_amdgcn_s_wait_asynccnt(0);
        }
        __syncthreads();                            // all waves' staging visible (+ sA on c==0)
        const float (*sWc)[DD] = sW[c & 1];
        #pragma unroll
        for (int k0 = 0; k0 < 32; k0 += 4) {
            v2f a, b;
            a[0] = sA[lm][ko + k0 + 2 * half + 0];
            a[1] = sA[lm][ko + k0 + 2 * half + 1];
            b[0] = sWc[k0 + 2 * half + 0][col0 + lm];
            b[1] = sWc[k0 + 2 * half + 1][col0 + lm];
            acc = __builtin_amdgcn_wmma_f32_16x16x4_f32(
                false, a, false, b, (short)0, acc, false, false);
        }
        __syncthreads();                            // buffer reuse guard
    }
#else
    for (int ko = 0; ko < DD; ko += 32) {
        __syncthreads();                            // also covers sA load on first pass
        const float4* Wv  = (const float4*)(W + (size_t)ko * DD);
        float4*       sWv = (float4*)&sW[0][0][0];
        for (int i = tid; i < 32 * DD / 4; i += 256) sWv[i] = Wv[i];
        __syncthreads();
        #pragma unroll
        for (int k0 = 0; k0 < 32; k0 += 4) {
            v2f a, b;
            a[0] = sA[lm][ko + k0 + 2 * half + 0];
            a[1] = sA[lm][ko + k0 + 2 * half + 1];
            b[0] = sW[0][k0 + 2 * half + 0][col0 + lm];
            b[1] = sW[0][k0 + 2 * half + 1][col0 + lm];
            acc = __builtin_amdgcn_wmma_f32_16x16x4_f32(
                false, a, false, b, (short)0, acc, false, false);
        }
    }
#endif
    // D layout: VGPR i -> M=i (lanes 0-15) / M=i+8 (lanes 16-31), N=lane%16
    #pragma unroll
    for (int i = 0; i < 8; ++i)
        xw[(size_t)(row0 + i + 8 * half) * DD + col0 + lm] = acc[i];
}

// ---------------- scalar GCN aggregation for score_s ----------------
__global__ void k_accs_init(const float* dinv, const float* xw_s, float* acc_s) {
    int i = blockIdx.x * 256 + threadIdx.x;
    if (i < NN) acc_s[i] = dinv[i] * dinv[i] * xw_s[i];
}
__global__ void k_accs_edges(const int* __restrict__ src, const int* __restrict__ dst,
                             const float* __restrict__ dinv, const float* __restrict__ xw_s,
                             float* __restrict__ acc_s) {
    int e = blockIdx.x * 256 + threadIdx.x;
    if (e >= EE) return;
    int s = src[e], d = dst[e];
    atomicAdd(&acc_s[d], dinv[s] * dinv[d] * xw_s[s]);
}
__global__ void k_score(const float* acc_s, const float* bs, const float* score_f, float* score) {
    int i = blockIdx.x * 256 + threadIdx.x;
    if (i < NN) score[i] = tanhf(ALPHA * (acc_s[i] + bs[0]) + (1.0f - ALPHA) * score_f[i]);
}

// ---------------- vector GCN aggregation for fusion conv ----------------
__global__ void k_xfu_init(const float* __restrict__ dinv, const float* __restrict__ xw_fu,
                           float* __restrict__ x_fused) {
    size_t i = (size_t)blockIdx.x * 256 + threadIdx.x;
    if (i < (size_t)NN * DD) {
        int n = (int)(i >> 7);
        x_fused[i] = dinv[n] * dinv[n] * xw_fu[i];
    }
}
// wave-per-edge; lane covers channels {lane, lane+32, lane+64, lane+96} -> coalesced 128B
__global__ __launch_bounds__(256) void k_agg_fu(const int* __restrict__ src, const int* __restrict__ dst,
                                                const float* __restrict__ dinv,
                                                const float* __restrict__ xw,
                                                float* __restrict__ outp) {
    int e = blockIdx.x * 8 + (threadIdx.x >> 5);
    int lane = threadIdx.x & 31;
    if (e >= EE) return;
    int s = src[e], d = dst[e];
    float norm = dinv[s] * dinv[d];
    const float* xs = xw + (size_t)s * DD;
    float* od = outp + (size_t)d * DD;
    #pragma unroll
    for (int j = 0; j < 4; ++j) {
        int c = lane + 32 * j;
        atomicAdd(&od[c], norm * xs[c]);
    }
}

// ---------------- per-graph top-k: bitonic sort of 1024 (score desc, idx asc) ----------------
__global__ __launch_bounds__(512) void k_topk(const float* __restrict__ score, int* __restrict__ perm) {
    __shared__ float key[1024];
    __shared__ int   idx[1024];
    const int b = blockIdx.x, tid = threadIdx.x;
    for (int i = tid; i < 1024; i += 512) {
        if (i < NPG) { key[i] = score[b * NPG + i]; idx[i] = i; }
        else         { key[i] = -3.402823466e38f;   idx[i] = i; }
    }
    __syncthreads();
    for (int ks = 2; ks <= 1024; ks <<= 1) {
        for (int j = ks >> 1; j > 0; j >>= 1) {
            for (int i = tid; i < 1024; i += 512) {
                int ixj = i ^ j;
                if (ixj > i) {
                    float ka = key[i], kb = key[ixj];
                    int   ia = idx[i], ib = idx[ixj];
                    bool before = (ka > kb) || (ka == kb && ia < ib);  // stable desc order
                    bool up = ((i & ks) == 0);
                    if (up ? !before : before) {
                        key[i] = kb; key[ixj] = ka;
                        idx[i] = ib; idx[ixj] = ia;
                    }
                }
            }
            __syncthreads();
        }
    }
    for (int i = tid; i < KSEL; i += 512)
        perm[b * KSEL + i] = b * NPG + idx[i];
}

// ---------------- node remap ----------------
__global__ void k_setmap(const int* perm, int* nodemap) {
    int i = blockIdx.x * 256 + threadIdx.x;
    if (i < NSEL) nodemap[perm[i]] = i;
}

// ---------------- order-preserving edge filter ----------------
__global__ __launch_bounds__(256) void k_edge_count(const int* __restrict__ src, const int* __restrict__ dst,
                                                    const int* __restrict__ nodemap, int* __restrict__ bcount) {
    __shared__ int cnt;
    if (threadIdx.x == 0) cnt = 0;
    __syncthreads();
    int e = blockIdx.x * 256 + threadIdx.x;
    int keep = 0;
    if (e < EE) keep = (nodemap[src[e]] >= 0 && nodemap[dst[e]] >= 0) ? 1 : 0;
    if (keep) atomicAdd(&cnt, 1);
    __syncthreads();
    if (threadIdx.x == 0) bcount[blockIdx.x] = cnt;
}
__global__ void k_scan(const int* bcount, int* boffs, int* mtot, int nb) {
    int run = 0;
    for (int i = 0; i < nb; ++i) { boffs[i] = run; run += bcount[i]; }
    mtot[0] = run;
}
__global__ __launch_bounds__(256) void k_edge_scatter(const int* __restrict__ src, const int* __restrict__ dst,
                                                      const int* __restrict__ nodemap,
                                                      const int* __restrict__ boffs,
                                                      const int* __restrict__ mtot,
                                                      float* __restrict__ out_e) {
    __shared__ int sf[256];
    const int tid = threadIdx.x;
    int e = blockIdx.x * 256 + tid;
    int r = -1, c = -1, keep = 0;
    if (e < EE) {
        r = nodemap[src[e]];
        c = nodemap[dst[e]];
        keep = (r >= 0 && c >= 0) ? 1 : 0;
    }
    sf[tid] = keep;
    __syncthreads();
    // Hillis-Steele inclusive scan
    for (int off = 1; off < 256; off <<= 1) {
        int v = (tid >= off) ? sf[tid - off] : 0;
        __syncthreads();
        sf[tid] += v;
        __syncthreads();
    }
    if (keep) {
        int pos = boffs[blockIdx.x] + sf[tid] - 1;
        int m = mtot[0];
        out_e[pos]     = (float)r;   // row 0: remapped src
        out_e[m + pos] = (float)c;   // row 1: remapped dst
    }
}

// ---------------- gather tail: x_out, batch_out, perm, x_ae ----------------
__global__ __launch_bounds__(128) void k_tail(const float* __restrict__ x_fused,
                                              const float* __restrict__ b_fu,
                                              const float* __restrict__ score,
                                              const int* __restrict__ perm,
                                              const int* __restrict__ batch,
                                              const int* __restrict__ mtot,
                                              float* __restrict__ out) {
    const int i = blockIdx.x;     // selected node 0..NSEL-1
    const int d = threadIdx.x;    // channel 0..127
    const int p = perm[i];
    const float s = score[p];
    const float v = x_fused[(size_t)p * DD + d] + b_fu[d];
    out[(size_t)i * DD + d] = v * s;                       // x_out
    const int m = mtot[0];
    const size_t base = XOUT_ELEMS + 2 * (size_t)m;        // after x_out + edge_index_new
    float* xae = out + base + 2 * (size_t)NSEL;            // after batch_out + perm
    xae[(size_t)i * DD + d] = v;                           // x_ae
    if (d == 0) {
        out[base + i]        = (float)batch[p];            // batch_out
        out[base + NSEL + i] = (float)p;                   // perm
    }
}

extern "C" void kernel_launch(void* const* d_in, const int* in_sizes, int n_in,
                              void* d_out, int out_size, void* d_ws, size_t ws_size,
                              hipStream_t stream) {
    const float* x    = (const float*)d_in[0];
    const int*   ei   = (const int*)d_in[1];
    const int*   srcp = ei;
    const int*   dstp = ei + EE;
    const int*   batch = (const int*)d_in[2];
    const float* W_s  = (const float*)d_in[3];
    const float* b_s  = (const float*)d_in[4];
    const float* W_f  = (const float*)d_in[5];
    const float* b_f  = (const float*)d_in[6];
    const float* W_fu = (const float*)d_in[7];
    const float* b_fu = (const float*)d_in[8];
    float* out = (float*)d_out;

    // workspace carve-up
    float* ws_f = (float*)d_ws;
    size_t off = 0;
    float* dinv    = ws_f + off; off += NN;               // degree -> dinv (in place)
    float* xw_s    = ws_f + off; off += NN;
    float* score_f = ws_f + off; off += NN;
    float* acc_s   = ws_f + off; off += NN;
    float* score   = ws_f + off; off += NN;
    float* xw_fu   = ws_f + off; off += (size_t)NN * DD;
    float* x_fused = ws_f + off; off += (size_t)NN * DD;
    int* permv   = (int*)(ws_f + off); off += NSEL;
    int* nodemap = (int*)(ws_f + off); off += NN;
    const int NBLK_E = (EE + 255) / 256;
    int* bcount  = (int*)(ws_f + off); off += NBLK_E;
    int* boffs   = (int*)(ws_f + off); off += NBLK_E;
    int* mtot    = (int*)(ws_f + off); off += 4;

    const int gN  = (NN + 255) / 256;
    const int gE  = (EE + 255) / 256;
    const int gND = (int)(((size_t)NN * DD + 255) / 256);

    // 1) degrees (edges + self loop) -> dinv
    k_fill_f  <<<gN, 256, 0, stream>>>(dinv, 1.0f, NN);
    k_deg_edges<<<gE, 256, 0, stream>>>(dstp, dinv);
    k_dinv    <<<gN, 256, 0, stream>>>(dinv);
    // 2) scalar projections
    k_scores  <<<NN / 8, 256, 0, stream>>>(x, W_s, W_f, b_f, xw_s, score_f);
    // 3) dense GEMM via f32 WMMA (async double-buffered W staging when available)
    k_gemm_fu <<<NN / 16, 256, 0, stream>>>(x, W_fu, xw_fu);
    // 4) structure score aggregation + fused score
    k_accs_init <<<gN, 256, 0, stream>>>(dinv, xw_s, acc_s);
    k_accs_edges<<<gE, 256, 0, stream>>>(srcp, dstp, dinv, xw_s, acc_s);
    k_score     <<<gN, 256, 0, stream>>>(acc_s, b_s, score_f, score);
    // 5) fusion conv aggregation
    k_xfu_init<<<gND, 256, 0, stream>>>(dinv, xw_fu, x_fused);
    k_agg_fu  <<<EE / 8, 256, 0, stream>>>(srcp, dstp, dinv, xw_fu, x_fused);
    // 6) per-graph top-k
    k_topk    <<<BB, 512, 0, stream>>>(score, permv);
    // 7) node remap
    k_fill_i  <<<gN, 256, 0, stream>>>(nodemap, -1, NN);
    k_setmap  <<<(NSEL + 255) / 256, 256, 0, stream>>>(permv, nodemap);
    // 8) order-preserving edge filter into d_out (after x_out region)
    k_edge_count  <<<NBLK_E, 256, 0, stream>>>(srcp, dstp, nodemap, bcount);
    k_scan        <<<1, 1, 0, stream>>>(bcount, boffs, mtot, NBLK_E);
    k_edge_scatter<<<NBLK_E, 256, 0, stream>>>(srcp, dstp, nodemap, boffs, mtot, out + XOUT_ELEMS);
    // 9) gather outputs (x_out, batch_out, perm, x_ae) at m-dependent offsets
    k_tail    <<<NSEL, 128, 0, stream>>>(x_fused, b_fu, score, permv, batch, mtot, out);
}